// detector_postprocess_7662221656277
// MI455X (gfx1250) — compile-verified
//
#include <hip/hip_runtime.h>
#include <stdint.h>

#define B_IMG 32
#define H_IMG 720
#define W_IMG 1280
#define IMG_PIX (H_IMG * W_IMG)   // 921600
#define KTOP 1024
#define CAP  2048
#define IOU_TH 0.1f

typedef float v2f __attribute__((ext_vector_type(2)));
typedef float v8f __attribute__((ext_vector_type(8)));
typedef int   v4i __attribute__((vector_size(16)));
typedef __attribute__((address_space(1))) v4i* gv4i_ptr;   // global int4*
typedef __attribute__((address_space(3))) v4i* lv4i_ptr;   // LDS int4*

// ---------------- kernel 0: zero output map + candidate counters ----------------
__global__ void k0_zero(float4* __restrict__ out4, long n4, int* __restrict__ counts) {
  long i = (long)blockIdx.x * blockDim.x + threadIdx.x;
  const long stride = (long)gridDim.x * blockDim.x;
  float4 z; z.x = 0.f; z.y = 0.f; z.z = 0.f; z.w = 0.f;
  for (; i < n4; i += stride) out4[i] = z;
  if (blockIdx.x == 0 && threadIdx.x < B_IMG) counts[threadIdx.x] = 0;
}

// ---------------- kernel 1: streaming threshold scan (async-to-LDS staged) ------
__global__ void k1_scan(const float* __restrict__ in, const float* __restrict__ thrP,
                        int* __restrict__ counts, unsigned long long* __restrict__ cands) {
  __shared__ __align__(16) float stage[256 * 4];
  const float thr = *thrP;
  const int tid = threadIdx.x;
  const long n4 = (long)B_IMG * IMG_PIX / 4;
  const long stride = (long)gridDim.x * blockDim.x;
  for (long i4 = (long)blockIdx.x * blockDim.x + tid; i4 < n4; i4 += stride) {
    const float* gp = in + i4 * 4;
    float4 v;
#if __has_builtin(__builtin_amdgcn_global_load_async_to_lds_b128)
    {
      // global generic ptrs are numerically AS1; low 32 bits of a flat LDS ptr are the LDS offset
      gv4i_ptr gsrc = (gv4i_ptr)(uintptr_t)(const void*)gp;
      lv4i_ptr ldst = (lv4i_ptr)(unsigned)(uintptr_t)(&stage[tid * 4]);
      __builtin_amdgcn_global_load_async_to_lds_b128(gsrc, ldst, 0, 0);
#if __has_builtin(__builtin_amdgcn_s_wait_asynccnt)
      __builtin_amdgcn_s_wait_asynccnt(0);
#else
      asm volatile("s_wait_asynccnt 0" ::: "memory");
#endif
      v = *(const float4*)&stage[tid * 4];
    }
#else
    v = *(const float4*)gp;
#endif
    const float vals[4] = {v.x, v.y, v.z, v.w};
    const long base = i4 * 4;
    const int img = (int)(base / IMG_PIX);
    const int fi  = (int)(base - (long)img * IMG_PIX);
#pragma unroll
    for (int e = 0; e < 4; ++e) {
      const float s = vals[e];
      if (s >= thr) {
        const int pos = atomicAdd(&counts[img], 1);
        if (pos < CAP) {
          const unsigned sb = __float_as_uint(s);
          // key: ascending == descending score, ties broken by ascending index (matches top_k)
          cands[(long)img * CAP + pos] =
              ((unsigned long long)(~sb) << 32) | (unsigned)(fi + e);
        }
      }
    }
  }
}

// ---------------- kernel 2: per-image sort + WMMA pairwise-IoU NMS + scatter ----
__global__ __launch_bounds__(1024) void k2_nms(const int* __restrict__ sizeP,
                                               const int* __restrict__ counts,
                                               const unsigned long long* __restrict__ cands,
                                               float* __restrict__ out) {
  // LDS layout (sort-key region overlaid on suppression-bitmask region):
  //   [0,4K)      scoreS   [4K,8K) idxS   [8K,12K) xsS   [12K,16K) ysS
  //   [16K,16K+128) keptS
  //   [16512, 16512+16K)  keys (2048 x u64)      -- dead after extraction
  //   [16512, 16512+128K) supS (1024 rows x 32 u32)
  constexpr int SMEM_HI = 16512;
  __shared__ __align__(16) unsigned char smem[SMEM_HI + KTOP * 32 * 4];
  float*    scoreS = (float*)(smem + 0);
  unsigned* idxS   = (unsigned*)(smem + 4096);
  float*    xsS    = (float*)(smem + 8192);
  float*    ysS    = (float*)(smem + 12288);
  unsigned* keptS  = (unsigned*)(smem + 16384);
  unsigned long long* keys = (unsigned long long*)(smem + SMEM_HI);
  unsigned* supS   = (unsigned*)(smem + SMEM_HI);

  const int img  = blockIdx.x;
  const int tid  = threadIdx.x;
  const int lane = tid & 31;
  const int wave = tid >> 5;

  const int cnt0 = counts[img];
  const int cnt  = cnt0 < CAP ? cnt0 : CAP;
  const int n    = cnt < KTOP ? cnt : KTOP;

  // ---- load candidate keys, pad to CAP ----
  for (int t = tid; t < CAP; t += 1024)
    keys[t] = (t < cnt) ? cands[(long)img * CAP + t] : ~0ull;
  __syncthreads();

  // ---- bitonic sort ascending over CAP=2048 keys ----
  for (unsigned k = 2; k <= CAP; k <<= 1) {
    for (unsigned j = k >> 1; j > 0; j >>= 1) {
      for (unsigned i = tid; i < CAP; i += 1024) {
        const unsigned ixj = i ^ j;
        if (ixj > i) {
          const unsigned long long a = keys[i], b = keys[ixj];
          const bool up = ((i & k) == 0);
          if ((a > b) == up) { keys[i] = b; keys[ixj] = a; }
        }
      }
      __syncthreads();
    }
  }

  // ---- extract top-K into per-candidate arrays ----
  {
    const unsigned long long kk = keys[tid];
    const unsigned idx = (unsigned)kk;
    const float sc = __uint_as_float(~(unsigned)(kk >> 32));
    const bool ok = (tid < n);
    scoreS[tid] = ok ? sc : 0.f;
    idxS[tid]   = ok ? idx : 0u;
    ysS[tid]    = ok ? (float)(idx / W_IMG) : 1.0e8f;
    xsS[tid]    = ok ? (float)(idx % W_IMG) : 1.0e8f;
  }
  __syncthreads();   // keys dead from here; supS may overwrite that region

  for (int w = tid; w < KTOP * 32; w += 1024) supS[w] = 0;
  if (tid < 32) keptS[tid] = 0;
  __syncthreads();

  // ---- build suppression bitmask with WMMA pairwise differences ----
  // dy tile = [y,1] x [1 ; -y^T]  via V_WMMA_F32_16X16X4_F32 (K slots 2,3 zero)
  const float fsize = (float)(*sizeP);
  const float area  = fsize * fsize;
  const float interTh = (IOU_TH * 2.0f * area) / (1.0f + IOU_TH);  // inter > th <=> iou > IOU_TH
  const int tilesN = (n + 15) >> 4;
  const int lsel = lane & 15;
  const bool lo = (lane < 16);

  for (int t_id = wave; t_id < 64 * 64; t_id += 32) {          // wave-uniform tile loop
    const int ti = t_id >> 6, tj = t_id & 63;
    if (tj < ti || ti >= tilesN || tj >= tilesN) continue;     // uniform per wave
    const int i0 = ti << 4, j0 = tj << 4;

    const float ya = ysS[i0 + lsel], xa = xsS[i0 + lsel];
    const float yb = ysS[j0 + lsel], xb = xsS[j0 + lsel];
    v2f aY, bY, aX, bX;
    aY[0] = lo ? ya : 0.f;   aY[1] = lo ? 1.f : 0.f;   // A: lanes0-15 K0,K1; lanes16-31 K2,K3=0
    bY[0] = lo ? 1.f : 0.f;  bY[1] = lo ? -yb : 0.f;   // B: row0 = ones, row1 = -y
    aX[0] = lo ? xa : 0.f;   aX[1] = lo ? 1.f : 0.f;
    bX[0] = lo ? 1.f : 0.f;  bX[1] = lo ? -xb : 0.f;
    v8f cz = {};
    v8f dy = __builtin_amdgcn_wmma_f32_16x16x4_f32(false, aY, false, bY, (short)0, cz, false, false);
    v8f dx = __builtin_amdgcn_wmma_f32_16x16x4_f32(false, aX, false, bX, (short)0, cz, false, false);

    const int jcol  = j0 + lsel;
    const int rbase = i0 + ((lane >> 4) << 3);
#pragma unroll
    for (int v = 0; v < 8; ++v) {
      const float iy = fmaxf(0.f, fsize - fabsf(dy[v]));
      const float ix = fmaxf(0.f, fsize - fabsf(dx[v]));
      const float inter = iy * ix;
      const int irow = rbase + v;
      if (inter > interTh && jcol > irow && irow < n && jcol < n)
        atomicOr(&supS[(irow << 5) + (jcol >> 5)], 1u << (jcol & 31));
    }
  }
  __syncthreads();

  // ---- greedy score-order reduction: one wave32 holds the 1024-bit removed mask ----
  if (tid < 32) {
    unsigned remv = 0, kept = 0;
    for (int i = 0; i < n; ++i) {
      const unsigned w = (unsigned)__builtin_amdgcn_readlane((int)remv, i >> 5);
      if (!((w >> (i & 31)) & 1u)) {            // uniform branch
        if (lane == (i >> 5)) kept |= (1u << (i & 31));
        remv |= supS[(i << 5) + lane];
      }
    }
    keptS[lane] = kept;
  }
  __syncthreads();

  // ---- scatter kept scores onto the pre-zeroed map ----
  if (tid < n && ((keptS[tid >> 5] >> (tid & 31)) & 1u))
    out[(long)img * IMG_PIX + idxS[tid]] = scoreS[tid];
}

extern "C" void kernel_launch(void* const* d_in, const int* in_sizes, int n_in,
                              void* d_out, int out_size, void* d_ws, size_t ws_size,
                              hipStream_t stream) {
  const float* pred = (const float*)d_in[0];
  const int*   sizeP = (const int*)d_in[1];
  const float* thrP  = (const float*)d_in[2];
  float* out = (float*)d_out;

  int* counts = (int*)d_ws;                                                  // 32 ints
  unsigned long long* cands = (unsigned long long*)((char*)d_ws + 256);      // 32*2048 u64

  const long n4 = (long)B_IMG * IMG_PIX / 4;
  k0_zero<<<4096, 256, 0, stream>>>((float4*)out, n4, counts);
  k1_scan<<<2048, 256, 0, stream>>>(pred, thrP, counts, cands);
  k2_nms<<<B_IMG, 1024, 0, stream>>>(sizeP, counts, cands, out);
}